// FrustumToVoxel_8169027797811
// MI455X (gfx1250) — compile-verified
//
#include <hip/hip_runtime.h>
#include <math.h>

// Problem constants (from reference setup_inputs)
#define ZDIM 16
#define YDIM 160
#define XDIM 160
#define NVIEW 6
#define CDIM 64
#define D0 48           // W  (grid[...,2] axis)
#define D1 32           // H  (grid[...,1] axis)
#define D2 88           // D  (grid[...,0] axis, contiguous)
#define CH (D0 * D1 * D2)        // floats per channel = 135168
#define FV (CDIM * CH)           // floats per view
#define GROUPS 8                 // channel groups per block
#define GC (CDIM / GROUPS)       // channels per group = 8
#define VOX 32                   // voxels per block (16 z x 2 y)

static __device__ __forceinline__ int iclamp(int v, int lo, int hi) {
  return v < lo ? lo : (v > hi ? hi : v);
}

__global__ __launch_bounds__(256)
void frustum2voxel_kernel(const float* __restrict__ feat,
                          const float* __restrict__ grid,
                          const unsigned char* __restrict__ mask,
                          float* __restrict__ out)
{
  // LDS tables laid out [corner][voxel] so phase-2 reads are conflict-free.
  __shared__ float sGrid[VOX][3];
  __shared__ int   sIdx[8][VOX];
  __shared__ float sW[8][VOX];
  __shared__ int   sFlag[VOX];

  const int tid = threadIdx.x;
  const int x   = blockIdx.x % XDIM;
  const int y0  = (blockIdx.x / XDIM) * 2;
  const int v   = tid & (VOX - 1);   // local voxel 0..31
  const int cg  = tid >> 5;          // channel group 0..7 (== wave id)
  const int z   = v & 15;
  const int y   = y0 + (v >> 4);

  float acc[GC];
#pragma unroll
  for (int j = 0; j < GC; ++j) acc[j] = 0.0f;

  for (int n = 0; n < NVIEW; ++n) {
    // ---- Phase 1: wave 0 computes per-voxel corner indices + weights ----
    if (tid < VOX) {
      const long voxoff = (((long)n * ZDIM + z) * YDIM + y) * XDIM + x;
      const float* gp = grid + voxoff * 3;

      // CDNA5 async global->LDS staging of the 3 grid coords (ASYNCcnt path).
      unsigned ldsOff = (unsigned)(unsigned long long)(const void*)&sGrid[v][0];
      unsigned long long ga = (unsigned long long)gp;
      asm volatile("global_load_async_to_lds_b32 %0, %1, off"
                   :: "v"(ldsOff), "v"(ga) : "memory");
      asm volatile("global_load_async_to_lds_b32 %0, %1, off offset:4"
                   :: "v"(ldsOff), "v"(ga) : "memory");
      asm volatile("global_load_async_to_lds_b32 %0, %1, off offset:8"
                   :: "v"(ldsOff), "v"(ga) : "memory");
      const unsigned m = mask[voxoff];   // overlaps with async transfer
      asm volatile("s_wait_asynccnt 0" ::: "memory");

      const float gx = sGrid[v][0];
      const float gy = sGrid[v][1];
      const float gz = sGrid[v][2];

      // align_corners=True mapping
      const float fx = (gx + 1.0f) * (0.5f * (float)(D2 - 1));
      const float fy = (gy + 1.0f) * (0.5f * (float)(D1 - 1));
      const float fz = (gz + 1.0f) * (0.5f * (float)(D0 - 1));
      const float xf = floorf(fx), yf = floorf(fy), zf = floorf(fz);
      const float tx = fx - xf, ty = fy - yf, tz = fz - zf;
      const int ix0 = (int)xf, iy0 = (int)yf, iz0 = (int)zf;

      float wx[2], wy[2], wz[2];
      int   cx[2], cy[2], cz[2];
      wx[0] = (ix0     >= 0 && ix0     < D2) ? (1.0f - tx) : 0.0f;
      wx[1] = (ix0 + 1 >= 0 && ix0 + 1 < D2) ? tx          : 0.0f;
      wy[0] = (iy0     >= 0 && iy0     < D1) ? (1.0f - ty) : 0.0f;
      wy[1] = (iy0 + 1 >= 0 && iy0 + 1 < D1) ? ty          : 0.0f;
      wz[0] = (iz0     >= 0 && iz0     < D0) ? (1.0f - tz) : 0.0f;
      wz[1] = (iz0 + 1 >= 0 && iz0 + 1 < D0) ? tz          : 0.0f;
      cx[0] = iclamp(ix0, 0, D2 - 1);  cx[1] = iclamp(ix0 + 1, 0, D2 - 1);
      cy[0] = iclamp(iy0, 0, D1 - 1);  cy[1] = iclamp(iy0 + 1, 0, D1 - 1);
      cz[0] = iclamp(iz0, 0, D0 - 1);  cz[1] = iclamp(iz0 + 1, 0, D0 - 1);

      // Skip masked voxels and fully out-of-bounds samples (saves gathers).
      const float ax = wx[0] + wx[1], ay = wy[0] + wy[1], az = wz[0] + wz[1];
      const int active = (m == 0) && (ax * ay * az > 0.0f);
      sFlag[v] = active;

#pragma unroll
      for (int k = 0; k < 8; ++k) {
        const int dx = k & 1, dy = (k >> 1) & 1, dz = k >> 2;
        sW[k][v]   = active ? (wz[dz] * wy[dy] * wx[dx]) : 0.0f;
        sIdx[k][v] = (cz[dz] * D1 + cy[dy]) * D2 + cx[dx];
      }
    }
    __syncthreads();

    // ---- Phase 2: all waves gather 8 corners x 8 channels ----
    if (sFlag[v]) {
      int   idx[8];
      float w[8];
#pragma unroll
      for (int k = 0; k < 8; ++k) { idx[k] = sIdx[k][v]; w[k] = sW[k][v]; }

      const float* __restrict__ fb =
          feat + (size_t)n * FV + (size_t)(cg * GC) * CH;
#pragma unroll
      for (int j = 0; j < GC; ++j) {
        const float* __restrict__ cp = fb + (size_t)j * CH;
        float s = 0.0f;
#pragma unroll
        for (int k = 0; k < 8; ++k) s = fmaf(w[k], cp[idx[k]], s);
        acc[j] += s;
      }
    }
    __syncthreads();
  }

  // ---- Epilogue: coalesced non-temporal stores ----
  // out index = ((c*X + x)*Y + y)*Z + z ; within a wave, lanes cover
  // (y0..y0+1) x (z 0..15) => 32 consecutive dwords per store.
#pragma unroll
  for (int j = 0; j < GC; ++j) {
    const int c = cg * GC + j;
    float* op = out + (((size_t)c * XDIM + x) * YDIM + y) * ZDIM + z;
    __builtin_nontemporal_store(acc[j], op);
  }
}

extern "C" void kernel_launch(void* const* d_in, const int* in_sizes, int n_in,
                              void* d_out, int out_size, void* d_ws, size_t ws_size,
                              hipStream_t stream) {
  (void)in_sizes; (void)n_in; (void)out_size; (void)d_ws; (void)ws_size;
  const float*         feat = (const float*)d_in[0];          // (1,6,64,48,32,88) f32
  const float*         grid = (const float*)d_in[1];          // (6,16,160,160,3)  f32
  const unsigned char* mask = (const unsigned char*)d_in[2];  // (6,16,160,160)    bool
  float* out = (float*)d_out;                                  // (1,64,160,160,16) f32

  const int nblocks = XDIM * (YDIM / 2);  // 12800 blocks, 256 threads (8 waves)
  frustum2voxel_kernel<<<nblocks, 256, 0, stream>>>(feat, grid, mask, out);
}